// TopKRouter_37623913513259
// MI455X (gfx1250) — compile-verified
//
#include <hip/hip_runtime.h>
#include <math.h>

// ---------------------------------------------------------------------------
// MoE top-2 router for MI455X (gfx1250, wave32).
//   logits = x @ W_r^T   (32768x2048) @ (2048x64)  -> fused softmax + top-2
// Memory-bound: 256 MB of x @ 23.3 TB/s ~= 11 us floor. We use
// V_WMMA_F32_16X16X4_F32 (f32 matches the reference's precision, so top-k
// ordering is preserved) and stage x through LDS in the A-fragment layout so
// the inner loop is ds_load_b64 + global_load_b64(W_r, L2-resident) + wmma.
// ---------------------------------------------------------------------------

typedef float v2f __attribute__((ext_vector_type(2)));
typedef float v4f __attribute__((ext_vector_type(4)));
typedef float v8f __attribute__((ext_vector_type(8)));

#define TOKENS  32768
#define DMODEL  2048
#define NEXP    64
#define TOPK    2

#define BLK_TOK 16              // tokens per block = one 16-row M tile
#define KC      128             // K chunk staged in LDS per iteration
#define NWAVES  4               // one wave per 16-expert N tile
#define THREADS (32 * NWAVES)   // 128 threads = 4 wave32

__global__ __launch_bounds__(THREADS)
void topk_router_kernel(const float* __restrict__ x,
                        const float* __restrict__ Wr,
                        float* __restrict__ gate_w,    // [TOKENS][2]
                        float* __restrict__ expert_i,  // [TOKENS][2] (float-valued)
                        float* __restrict__ probs)     // [TOKENS][NEXP]
{
    // A fragments pre-swizzled: ldsA[kk][lane] = float2 this lane feeds to
    // WMMA step kk.  ISA A-layout (16x4 f32): lanes 0-15 hold M=lane,K={0,1};
    // lanes 16-31 hold M=lane-16,K={2,3}.
    __shared__ v2f   ldsA[KC / 4][32];            // 8 KB
    __shared__ float ldsL[BLK_TOK][NEXP];         // 4 KB logits
    __shared__ float sMax[BLK_TOK], sSum[BLK_TOK];

    const int tid  = threadIdx.x;
    const int lane = tid & 31;
    const int wave = tid >> 5;
    const int tok0 = blockIdx.x * BLK_TOK;

    const int half = lane >> 4;     // 0 -> K pair {0,1}, 1 -> K pair {2,3}
    const int mrow = lane & 15;     // M row (A) / N col (B) for this lane
    const int ebase = wave * 16;    // this wave's expert tile

    v8f acc = {};                   // 16x16 f32 C/D tile (8 VGPRs)

    // B fragment source: W_r is (E, D) row-major; B[k][n] = W_r[ebase+n][k].
    // Lane `mrow` streams expert row (ebase+mrow); per step it needs the
    // K pair {k, k+1} (lanes 0-15) or {k+2, k+3} (lanes 16-31).
    const float* wrow = Wr + (size_t)(ebase + mrow) * DMODEL + 2 * half;

    for (int k0 = 0; k0 < DMODEL; k0 += KC) {
        // ---- cooperative stage of x[tok0..+16)[k0..+KC) into LDS, swizzled.
        // Thread t loads float4s of token (t>>3); one float4 covering K=4j..4j+3
        // is exactly one fragment: elems {0,1} -> lane m, elems {2,3} -> lane m+16.
        {
            const int t  = tid >> 3;
            const int jb = tid & 7;
            const float* xrow = x + (size_t)(tok0 + t) * DMODEL + k0;
            #pragma unroll
            for (int i = 0; i < (KC / 4) / 8; ++i) {
                int j = jb + 8 * i;
                v4f v = *(const v4f*)(xrow + 4 * j);   // global_load_b128
                v2f lo = { v.x, v.y };
                v2f hi = { v.z, v.w };
                ldsA[j][t]      = lo;                  // ds_store_b64
                ldsA[j][t + 16] = hi;
            }
        }
        __syncthreads();

        // ---- KC/4 WMMA steps: D = A(16x4) * B(4x16) + C
        const float* wk = wrow + k0;
        #pragma unroll 8
        for (int kk = 0; kk < KC / 4; ++kk) {
            v2f a = ldsA[kk][lane];                    // ds_load_b64
            v2f b = *(const v2f*)(wk + 4 * kk);        // global_load_b64 (L2 hit)
            acc = __builtin_amdgcn_wmma_f32_16x16x4_f32(
                      false, a, false, b, (short)0, acc, false, false);
        }
        __syncthreads();
    }

    // ---- scatter this wave's 16x16 logit tile to LDS.
    // C/D layout: VGPR r, lane L -> M = r + 8*(L>=16), N = L&15.
    {
        const int n = ebase + mrow;
        #pragma unroll
        for (int r = 0; r < 8; ++r) {
            ldsL[r + 8 * half][n] = acc[r];
        }
    }
    __syncthreads();

    // ---- per-token softmax stats + top-2 (16 lanes of wave 0).
    // softmax is monotonic in logits, so top-2 on logits == top-2 on probs;
    // strict '>' keeps the lower index on ties, matching jax.lax.top_k.
    if (tid < BLK_TOK) {
        const int m = tid;
        float mx = -INFINITY;
        for (int e = 0; e < NEXP; ++e) mx = fmaxf(mx, ldsL[m][e]);
        float sum = 0.0f;
        float b1 = -INFINITY, b2 = -INFINITY;
        int   i1 = 0, i2 = 0;
        for (int e = 0; e < NEXP; ++e) {
            float l = ldsL[m][e];
            sum += expf(l - mx);
            if (l > b1)      { b2 = b1; i2 = i1; b1 = l; i1 = e; }
            else if (l > b2) { b2 = l;  i2 = e; }
        }
        sMax[m] = mx;
        sSum[m] = sum;
        // gate renorm: softmax denominator cancels.
        float p1 = expf(b1 - mx), p2 = expf(b2 - mx);
        float inv = 1.0f / (p1 + p2);
        size_t t = (size_t)(tok0 + m);
        gate_w[2 * t]     = p1 * inv;
        gate_w[2 * t + 1] = p2 * inv;
        expert_i[2 * t]     = (float)i1;
        expert_i[2 * t + 1] = (float)i2;
    }
    __syncthreads();

    // ---- coalesced router_probs writes: 16*64 values / 128 threads = 8 each.
    #pragma unroll
    for (int i = 0; i < (BLK_TOK * NEXP) / THREADS; ++i) {
        int f = tid + THREADS * i;
        int m = f >> 6;
        int e = f & 63;
        probs[(size_t)(tok0 + m) * NEXP + e] =
            expf(ldsL[m][e] - sMax[m]) / sSum[m];
    }
}

extern "C" void kernel_launch(void* const* d_in, const int* in_sizes, int n_in,
                              void* d_out, int out_size, void* d_ws, size_t ws_size,
                              hipStream_t stream) {
    (void)in_sizes; (void)n_in; (void)out_size; (void)d_ws; (void)ws_size;
    const float* x  = (const float*)d_in[0];   // (32768, 2048) f32
    const float* Wr = (const float*)d_in[1];   // (64, 2048) f32

    float* out    = (float*)d_out;
    float* gate   = out;                               // 32768*2
    float* eidx   = out + (size_t)TOKENS * TOPK;       // 32768*2
    float* probs  = out + (size_t)2 * TOKENS * TOPK;   // 32768*64

    dim3 grid(TOKENS / BLK_TOK);   // 2048 blocks
    dim3 block(THREADS);           // 128 threads = 4 wave32
    topk_router_kernel<<<grid, block, 0, stream>>>(x, Wr, gate, eidx, probs);
}